// GateRow_4286377361935
// MI455X (gfx1250) — compile-verified
//
#include <hip/hip_runtime.h>
#include <cstdint>

// Problem constants (match reference)
#define B_ROWS  16384
#define N_IN    8192
#define N_GATES 8192

// Tiling
#define ROWS    16            // rows of x staged in LDS per block (16*8192 = 128 KB)
#define THREADS 256           // 8 waves (wave32)
#define GPT     4             // gates per thread per chunk -> float4 NT store
#define CHUNK   (THREADS*GPT) // 1024 gates per chunk, 8 chunks cover N_GATES

typedef __attribute__((ext_vector_type(4))) unsigned int v4u;
typedef __attribute__((ext_vector_type(8))) int          v8i;
typedef __attribute__((ext_vector_type(4))) int          v4i;
typedef __attribute__((ext_vector_type(4))) float        v4f;

__global__ __launch_bounds__(THREADS)
void gate_fwd_kernel(const unsigned char* __restrict__ x,      // (B, N_IN) bool bytes
                     const unsigned char* __restrict__ gates,  // (N_GATES, 4) bool bytes
                     const int*           __restrict__ choices,// (N_GATES, 2) int32
                     float*               __restrict__ out)    // (B, N_GATES) float
{
    extern __shared__ unsigned char xs[]; // ROWS * N_IN bytes of x tile

    const int b0 = blockIdx.x * ROWS;

#if __has_builtin(__builtin_amdgcn_tensor_load_to_lds) && __has_builtin(__builtin_amdgcn_s_wait_tensorcnt)
    // --- TDM: one tensor_load_to_lds moves the whole 16x8192B x-tile into LDS ---
    // Wave-uniform branch so only wave 0 issues the TDM op (TDM ignores EXEC).
    if (threadIdx.x < 32) {
        const unsigned long long ga =
            (unsigned long long)(uintptr_t)(x + (size_t)b0 * N_IN);
        const unsigned lds_off = (unsigned)(uintptr_t)xs; // low 32 bits of flat LDS addr = LDS offset

        // D# group 0 (128b): count=1 | lds_addr | global_addr[56:0] | type=2
        v4u g0;
        g0[0] = 1u;                                        // count=1, is_restore=0, gather off
        g0[1] = lds_off;                                   // lds_addr (bytes)
        g0[2] = (unsigned)ga;                              // global_addr[31:0]
        g0[3] = (unsigned)((ga >> 32) & 0x01FFFFFFu)       // global_addr[56:32]
              | (2u << 30);                                // type = 2 ("image")

        // D# group 1 (256b): data_size=1B, tensor_dim0=N_IN, tensor_dim1=B,
        // tile_dim0=N_IN, tile_dim1=ROWS, tensor_dim0_stride=N_IN
        v8i g1;
        g1[0] = 0;                                                   // wg_mask=0, data_size=0(1B), no pad/iter
        g1[1] = (int)(((unsigned)N_IN & 0xFFFFu) << 16);             // tensor_dim0[15:0] << 16
        g1[2] = (int)((((unsigned)N_IN >> 16) & 0xFFFFu)
              |       (((unsigned)B_ROWS & 0xFFFFu) << 16));         // tensor_dim0[31:16] | tensor_dim1[15:0]<<16
        g1[3] = (int)((((unsigned)B_ROWS >> 16) & 0xFFFFu)
              |       (((unsigned)N_IN & 0xFFFFu) << 16));           // tensor_dim1[31:16] | tile_dim0<<16
        g1[4] = ROWS;                                                // tile_dim1 (tile_dim2=0 unused)
        g1[5] = N_IN;                                                // tensor_dim0_stride[31:0]
        g1[6] = 0;                                                   // stride0[47:32] | dim1_stride[15:0]
        g1[7] = 0;

        const v4i z4 = {0, 0, 0, 0};                    // groups 2/3 unused (2-D tile)
        const v8i z8 = {0, 0, 0, 0, 0, 0, 0, 0};        // trailing descriptor arg (clang-23 6-arg form)
        __builtin_amdgcn_tensor_load_to_lds(g0, g1, z4, z4, z8, 0);
        __builtin_amdgcn_s_wait_tensorcnt(0);
    }
#else
    // Fallback: cooperative vectorized copy
    {
        const uint4* src = (const uint4*)(x + (size_t)b0 * N_IN);
        uint4* dst = (uint4*)xs;
        for (int i = threadIdx.x; i < (ROWS * N_IN) / 16; i += THREADS)
            dst[i] = src[i];
    }
#endif
    __syncthreads();

    const int gbase = threadIdx.x * GPT;

    for (int c = 0; c < N_GATES; c += CHUNK) {
        const int g = c + gbase; // 4 consecutive gates for this thread

        // choices[g..g+3]: 8 int32 = two aligned int4 loads (L2-resident)
        const int4* chp = (const int4*)(choices + 2 * g);
        const int4 cA = chp[0];
        const int4 cB = chp[1];
        const int c0[GPT] = {cA.x, cA.z, cB.x, cB.z};
        const int c1[GPT] = {cA.y, cA.w, cB.y, cB.w};

        // gates[g..g+3][0..3]: 16 bool bytes = one aligned b128 load;
        // each uint holds the 4-entry LUT for one gate (byte j = gates[g][j])
        const uint4 gw4 = *(const uint4*)(gates + 4 * g);
        const unsigned lut[GPT] = {gw4.x, gw4.y, gw4.z, gw4.w};

        #pragma unroll 4
        for (int r = 0; r < ROWS; ++r) {
            const unsigned char* xr = xs + r * N_IN;

            v4f o;
            #pragma unroll
            for (int i = 0; i < GPT; ++i) {
                const unsigned a   = (unsigned)xr[c0[i]];      // ds_load_u8 gather
                const unsigned b   = (unsigned)xr[c1[i]];      // ds_load_u8 gather
                const unsigned idx = (a << 1) | b;             // 0..3
                o[i] = (float)((lut[i] >> (idx << 3)) & 1u);   // LUT byte extract
            }

            // coalesced non-temporal b128 store (output never re-read)
            v4f* outp = (v4f*)(out + (size_t)(b0 + r) * N_GATES + g);
            __builtin_nontemporal_store(o, outp);
        }
    }
}

extern "C" void kernel_launch(void* const* d_in, const int* in_sizes, int n_in,
                              void* d_out, int out_size, void* d_ws, size_t ws_size,
                              hipStream_t stream) {
    const unsigned char* x       = (const unsigned char*)d_in[0]; // bool bytes
    const unsigned char* gates   = (const unsigned char*)d_in[1]; // bool bytes
    const int*           choices = (const int*)d_in[2];           // int32
    float*               out     = (float*)d_out;

    const size_t shmem = (size_t)ROWS * N_IN; // 128 KB dynamic LDS (WGP has 320 KB)
    (void)hipFuncSetAttribute((const void*)gate_fwd_kernel,
                              hipFuncAttributeMaxDynamicSharedMemorySize,
                              (int)shmem);

    dim3 grid(B_ROWS / ROWS); // 1024 blocks
    dim3 block(THREADS);
    gate_fwd_kernel<<<grid, block, shmem, stream>>>(x, gates, choices, out);
}